// HashGridEncoder_25821343383805
// MI455X (gfx1250) — compile-verified
//
#include <hip/hip_runtime.h>
#include <math.h>
#include <stdint.h>

#define N_LEVELS 16
#define LOG2_T 15
#define TSIZE (1 << LOG2_T)
#define TMASK (TSIZE - 1)
#define PTS_PER_BLOCK 512
#define BLOCK 256
#define ITERS (PTS_PER_BLOCK / 16) // 32
#define TILE_ELEMS (PTS_PER_BLOCK * 3) // 1536 floats per block

typedef float    v2f __attribute__((ext_vector_type(2)));
typedef unsigned v4u __attribute__((ext_vector_type(4)));
typedef int      v4i __attribute__((ext_vector_type(4)));
typedef int      v8i __attribute__((ext_vector_type(8)));

struct Res16 { float r[N_LEVELS]; };

__global__ __launch_bounds__(BLOCK) void hashgrid_fwd(
    const float* __restrict__ x, const float* __restrict__ mn,
    const float* __restrict__ mx, const float* __restrict__ table,
    float* __restrict__ out, int N, Res16 rt)
{
    __shared__ float sx[TILE_ELEMS]; // 6 KB staged coords

    const int tid  = threadIdx.x;
    const int lvl  = tid & 15;
    const int slot = tid >> 4;
    const int B0   = blockIdx.x * PTS_PER_BLOCK;

    // ---- Stage this block's x tile into LDS with one TDM descriptor (wave 0).
    // OOB reads return zero (ISA 8.2), so tensor_dim0 = remaining elements
    // handles the tail block with no clamping code.
    if (tid < 32) {
        const unsigned ldsoff = (unsigned)(uintptr_t)&sx[0]; // LDS byte offset
        const unsigned long ga = (unsigned long)(uintptr_t)x
                               + (unsigned long)B0 * 12ul;   // tile start (bytes)
        const unsigned rem = (unsigned)((long)N * 3 - (long)B0 * 3);

        v4u g0;
        g0[0] = 1u;                                      // count=1 (valid), user mode
        g0[1] = ldsoff;                                  // lds_addr
        g0[2] = (unsigned)(ga & 0xffffffffu);            // global_addr[31:0]
        g0[3] = (unsigned)((ga >> 32) & 0x01ffffffu)     // global_addr[56:32]
              | 0x80000000u;                             // type=2 ("image")

        v8i g1;
        g1[0] = (int)(2u << 16);                         // data_size=4B, mask=0
        g1[1] = (int)((rem & 0xffffu) << 16);            // tensor_dim0[15:0]
        g1[2] = (int)((rem >> 16) | (1u << 16));         // tensor_dim0[31:16], tensor_dim1=1
        g1[3] = (int)((unsigned)TILE_ELEMS << 16);       // tile_dim0=1536
        g1[4] = 0;                                       // tile_dim1=0 (1-D), tile_dim2=0
        g1[5] = (int)rem;                                // tensor_dim0_stride (unused for 1-D)
        g1[6] = 0;
        g1[7] = 0;

        v4i z4 = {0, 0, 0, 0};
        v8i z8 = {0, 0, 0, 0, 0, 0, 0, 0};
        __builtin_amdgcn_tensor_load_to_lds(g0, g1, z4, z4, z8, 0);
        __builtin_amdgcn_s_wait_tensorcnt(0);
    }
    __syncthreads();

    // ---- Per-thread (level-uniform) constants
    const float res = rt.r[lvl];
    const float mn0 = mn[0], mn1 = mn[1], mn2 = mn[2];
    const float i0 = 1.0f / (mx[0] - mn0);
    const float i1 = 1.0f / (mx[1] - mn1);
    const float i2 = 1.0f / (mx[2] - mn2);
    const float* __restrict__ tl = table + (size_t)lvl * (TSIZE * 2);

    #pragma unroll 2
    for (int it = 0; it < ITERS; ++it) {
        const int pi = it * 16 + slot;
        const int p  = B0 + pi;

        const float px = ((sx[pi * 3 + 0] - mn0) * i0) * res;
        const float py = ((sx[pi * 3 + 1] - mn1) * i1) * res;
        const float pz = ((sx[pi * 3 + 2] - mn2) * i2) * res;

        const float fx = floorf(px), fy = floorf(py), fz = floorf(pz);
        const float wx = px - fx,   wy = py - fy,   wz = pz - fz;
        const int   ix = (int)fx,   iy = (int)fy,   iz = (int)fz;

        // Spatial hash: per-axis prime products, XOR-combined per corner.
        const uint32_t hx0 = (uint32_t)ix;
        const uint32_t hx1 = (uint32_t)(ix + 1);
        const uint32_t hy0 = (uint32_t)iy       * 2654435761u;
        const uint32_t hy1 = (uint32_t)(iy + 1) * 2654435761u;
        const uint32_t hz0 = (uint32_t)iz       * 805459861u;
        const uint32_t hz1 = (uint32_t)(iz + 1) * 805459861u;
        const uint32_t e00 = hx0 ^ hy0, e10 = hx1 ^ hy0;
        const uint32_t e01 = hx0 ^ hy1, e11 = hx1 ^ hy1;

        // 8 unconditional gathers (mask keeps indices in-bounds) so loads pipeline.
        const v2f f0 = *(const v2f*)(tl + 2 * ((e00 ^ hz0) & TMASK)); // (0,0,0)
        const v2f f1 = *(const v2f*)(tl + 2 * ((e10 ^ hz0) & TMASK)); // (1,0,0)
        const v2f f2 = *(const v2f*)(tl + 2 * ((e01 ^ hz0) & TMASK)); // (0,1,0)
        const v2f f3 = *(const v2f*)(tl + 2 * ((e11 ^ hz0) & TMASK)); // (1,1,0)
        const v2f f4 = *(const v2f*)(tl + 2 * ((e00 ^ hz1) & TMASK)); // (0,0,1)
        const v2f f5 = *(const v2f*)(tl + 2 * ((e10 ^ hz1) & TMASK)); // (1,0,1)
        const v2f f6 = *(const v2f*)(tl + 2 * ((e01 ^ hz1) & TMASK)); // (0,1,1)
        const v2f f7 = *(const v2f*)(tl + 2 * ((e11 ^ hz1) & TMASK)); // (1,1,1)

        const float wx0 = 1.0f - wx, wy0 = 1.0f - wy, wz0 = 1.0f - wz;
        const float a = wx0 * wy0, b = wx * wy0, c = wx0 * wy, d = wx * wy;

        v2f acc = f0 * (a * wz0);
        acc += f1 * (b * wz0);
        acc += f2 * (c * wz0);
        acc += f3 * (d * wz0);
        acc += f4 * (a * wz);
        acc += f5 * (b * wz);
        acc += f6 * (c * wz);
        acc += f7 * (d * wz);

        if (p < N) {
            // Contiguous across the 16 level-lanes -> full 128B lines; stream past L2.
            __builtin_nontemporal_store(acc, (v2f*)(out + (size_t)p * 32 + lvl * 2));
        }
    }
}

extern "C" void kernel_launch(void* const* d_in, const int* in_sizes, int n_in,
                              void* d_out, int out_size, void* d_ws, size_t ws_size,
                              hipStream_t stream) {
    const float* x  = (const float*)d_in[0];
    const float* mn = (const float*)d_in[1];
    const float* mx = (const float*)d_in[2];
    const float* tb = (const float*)d_in[3];
    float* out = (float*)d_out;
    const int N = in_sizes[0] / 3;
    if (N <= 0) return;

    // Per-level resolutions, identical float64 math to the numpy reference
    // (values near 16*2^k are rounding-sensitive -> compute, don't hardcode).
    Res16 rt;
    const double b = pow(512.0 / 16.0, 1.0 / 15.0);
    for (int l = 0; l < N_LEVELS; ++l)
        rt.r[l] = (float)floor(16.0 * pow(b, (double)l));

    const int grid = (N + PTS_PER_BLOCK - 1) / PTS_PER_BLOCK;
    hashgrid_fwd<<<grid, BLOCK, 0, stream>>>(x, mn, mx, tb, out, N, rt);
}